// KRISPMedVQAModel_33122787786758
// MI455X (gfx1250) — compile-verified
//
#include <hip/hip_runtime.h>
#include <math.h>

#define B_   128
#define N_   65536
#define E_   1048576
#define CLIP 768
#define PD   512

typedef float v2f __attribute__((ext_vector_type(2)));
typedef float v8f __attribute__((ext_vector_type(8)));

// ---------------- init: deg=1 (self loops), zero pooled sums / counts ----------------
__global__ void k_init(float* __restrict__ deg, float* __restrict__ sums,
                       float* __restrict__ cnts) {
    int i = blockIdx.x * 256 + threadIdx.x;           // grid covers N_
    if (i < N_)     deg[i]  = 1.0f;
    if (i < B_*PD)  sums[i] = 0.0f;
    if (i < B_)     cnts[i] = 0.0f;
}

// ---------------- degree: deg[dst] += 1 over all edges ----------------
__global__ void k_deg(const int* __restrict__ ei, float* __restrict__ deg) {
    int e = blockIdx.x * 256 + threadIdx.x;
    if (e < E_) unsafeAtomicAdd(&deg[ei[E_ + e]], 1.0f);
}

// ---------------- self-loop term: agg[i] = x[i] / deg[i] (full overwrite) ----------------
__global__ void k_selfloop(const float* __restrict__ x, const float* __restrict__ deg,
                           float* __restrict__ agg) {
    size_t i = (size_t)blockIdx.x * 256 + threadIdx.x;   // < N_*PD
    int row = (int)(i >> 9);
    agg[i] = x[i] * (1.0f / deg[row]);                   // dis[i]*dis[i] = 1/deg
}

// ---------------- edge scatter: agg[dst] += x[src] * rsqrt(deg[s])*rsqrt(deg[d]) ----------------
__global__ void __launch_bounds__(256)
k_edges(const int* __restrict__ ei, const float* __restrict__ x,
        const float* __restrict__ deg, float* __restrict__ agg) {
    int wave = (int)((blockIdx.x * 256 + threadIdx.x) >> 5);   // one wave32 per edge
    int lane = threadIdx.x & 31;
    int src = ei[wave];
    int dst = ei[E_ + wave];
    float nrm = __frsqrt_rn(deg[src]) * __frsqrt_rn(deg[dst]);
    const float* xs = x   + (size_t)src * PD;
    float*       ad = agg + (size_t)dst * PD;
#pragma unroll
    for (int i = 0; i < 16; ++i) {                 // 512 cols, coalesced per lane
        int c = lane + (i << 5);
        unsafeAtomicAdd(&ad[c], xs[c] * nrm);
    }
}

// ---------------- node counts per batch ----------------
__global__ void k_cnts(const int* __restrict__ bv, float* __restrict__ cnts) {
    int i = blockIdx.x * 256 + threadIdx.x;
    if (i < N_) unsafeAtomicAdd(&cnts[bv[i]], 1.0f);
}

// ---------------- fused WMMA GEMM (agg @ Wg + bg) -> ReLU -> mean-pool atomic sums ----------------
__global__ void __launch_bounds__(256)
k_gcn_gemm_pool(const float* __restrict__ agg, const float* __restrict__ Wg,
                const float* __restrict__ bg, const int* __restrict__ bv,
                float* __restrict__ sums) {
    // one wave per 16x16 output tile; exact coverage -> EXEC all ones for WMMA
    int wave  = (int)((blockIdx.x * 256 + threadIdx.x) >> 5);
    int lane  = threadIdx.x & 31;
    int tileM = wave >> 5;            // N_/16 = 4096 row tiles
    int tileN = wave & 31;            // PD/16 = 32 col tiles
    int half  = lane >> 4;            // K-half selector per ISA A/B layout
    int l15   = lane & 15;
    int arow_m = tileM * 16 + l15;    // A: both lane halves hold M=0..15
    int col    = tileN * 16 + l15;    // B/C/D: N = lane & 15
    int koff   = half * 2;            // lanes16-31 carry K+2/K+3

    const float* arow = agg + (size_t)arow_m * PD + koff;
    const float* bcol = Wg  + (size_t)koff * PD + col;

    v8f c = {};
    for (int k = 0; k < PD; k += 4) {
        float2 av = *(const float2*)(arow + k);      // A[m][k+koff], A[m][k+koff+1]
        v2f a; a.x = av.x; a.y = av.y;
        v2f b;
        b.x = bcol[(size_t)k * PD];                  // Wg[k+koff  ][col]
        b.y = bcol[(size_t)(k + 1) * PD];            // Wg[k+koff+1][col]
        c = __builtin_amdgcn_wmma_f32_16x16x4_f32(
                /*neg_a=*/false, a, /*neg_b=*/false, b,
                /*c_mod=*/(short)0, c, /*reuse_a=*/false, /*reuse_b=*/false);
    }

    float bgv   = bg[col];
    int   rbase = tileM * 16 + half * 8;             // D layout: M = j (+8 for upper half)
#pragma unroll
    for (int j = 0; j < 8; ++j) {
        float v = c[j] + bgv;
        v = v > 0.0f ? v : 0.0f;                     // ReLU
        int r = rbase + j;
        unsafeAtomicAdd(&sums[(size_t)bv[r] * PD + col], v);   // pool into batch sums
    }
}

// ---------------- implicit branch: [img | txt] projections -> combined [B, 1024] ----------------
__global__ void k_proj(const float* __restrict__ imgF, const float* __restrict__ txtF,
                       const float* __restrict__ Wi, const float* __restrict__ bi,
                       const float* __restrict__ Wt, const float* __restrict__ bt,
                       float* __restrict__ comb) {
    int i = blockIdx.x * 256 + threadIdx.x;          // < B_*1024
    int b = i >> 10;
    int j = i & 1023;
    const float *W, *bias, *f; int jj;
    if (j < PD) { W = Wi; bias = bi; f = imgF + (size_t)b * CLIP; jj = j; }
    else        { W = Wt; bias = bt; f = txtF + (size_t)b * CLIP; jj = j - PD; }
    float acc = bias[jj];
    for (int k = 0; k < CLIP; ++k) acc += f[k] * W[(size_t)k * PD + jj];
    comb[i] = acc;
}

// ---------------- h1 = relu(combined @ W1 + b1) ----------------
__global__ void k_h1(const float* __restrict__ comb, const float* __restrict__ W1,
                     const float* __restrict__ b1, float* __restrict__ h1) {
    int i = blockIdx.x * 256 + threadIdx.x;          // < B_*PD
    int b = i >> 9;
    int j = i & 511;
    float acc = b1[j];
    const float* c = comb + (size_t)b * (2 * PD);
    for (int k = 0; k < 2 * PD; ++k) acc += c[k] * W1[(size_t)k * PD + j];
    h1[i] = acc > 0.0f ? acc : 0.0f;
}

// ---------------- y = h @ w + b  (PD -> 1 head, per batch row) ----------------
__global__ void k_vec1(const float* __restrict__ h, const float* __restrict__ w,
                       const float* __restrict__ b0, float* __restrict__ y) {
    int b = threadIdx.x;                              // 128 threads
    if (b >= B_) return;
    float acc = b0[0];
    const float* hb = h + (size_t)b * PD;
    for (int k = 0; k < PD; ++k) acc += hb[k] * w[k];
    y[b] = acc;
}

// ---------------- hs = relu((sums/cnt) @ Ws1 + bs1) ----------------
__global__ void k_hs(const float* __restrict__ sums, const float* __restrict__ cnts,
                     const float* __restrict__ Ws1, const float* __restrict__ bs1,
                     float* __restrict__ hs) {
    int i = blockIdx.x * 256 + threadIdx.x;          // < B_*PD
    int b = i >> 9;
    int j = i & 511;
    float inv = 1.0f / fmaxf(cnts[b], 1.0f);
    const float* s = sums + (size_t)b * PD;
    float dot = 0.0f;
    for (int k = 0; k < PD; ++k) dot += s[k] * Ws1[(size_t)k * PD + j];
    float acc = bs1[j] + dot * inv;
    hs[i] = acc > 0.0f ? acc : 0.0f;
}

// ---------------- logits = max(yimp, ysym); loss = mean BCEWithLogits ----------------
__global__ void k_final(const float* __restrict__ yimp, const float* __restrict__ ysym,
                        const float* __restrict__ ans, float* __restrict__ out) {
    __shared__ float red[128];
    int i = threadIdx.x;                              // 128 threads
    float l = fmaxf(yimp[i], ysym[i]);
    out[i] = l;
    // logaddexp(0, l) - l*y  == max(l,0) + log1p(exp(-|l|)) - l*y
    red[i] = fmaxf(l, 0.0f) + log1pf(expf(-fabsf(l))) - l * ans[i];
    __syncthreads();
    for (int s = 64; s > 0; s >>= 1) {
        if (i < s) red[i] += red[i + s];
        __syncthreads();
    }
    if (i == 0) out[B_] = red[0] * (1.0f / (float)B_);
}

extern "C" void kernel_launch(void* const* d_in, const int* in_sizes, int n_in,
                              void* d_out, int out_size, void* d_ws, size_t ws_size,
                              hipStream_t stream) {
    const float* imgF = (const float*)d_in[0];
    const float* txtF = (const float*)d_in[1];
    const float* ans  = (const float*)d_in[2];
    const float* x    = (const float*)d_in[3];
    const int*   ei   = (const int*)  d_in[4];
    const int*   bv   = (const int*)  d_in[5];
    const float* Wi   = (const float*)d_in[6];
    const float* bi   = (const float*)d_in[7];
    const float* Wt   = (const float*)d_in[8];
    const float* bt   = (const float*)d_in[9];
    const float* W1   = (const float*)d_in[10];
    const float* b1   = (const float*)d_in[11];
    const float* W2   = (const float*)d_in[12];
    const float* b2   = (const float*)d_in[13];
    const float* Wg   = (const float*)d_in[14];
    const float* bg   = (const float*)d_in[15];
    const float* Ws1  = (const float*)d_in[16];
    const float* bs1  = (const float*)d_in[17];
    const float* Ws2  = (const float*)d_in[18];
    const float* bs2  = (const float*)d_in[19];
    (void)in_sizes; (void)n_in; (void)out_size; (void)ws_size;

    float* ws   = (float*)d_ws;
    float* agg  = ws;  ws += (size_t)N_ * PD;   // 128 MB: A_norm @ X
    float* deg  = ws;  ws += N_;
    float* sums = ws;  ws += B_ * PD;
    float* cnts = ws;  ws += B_;
    float* comb = ws;  ws += B_ * 2 * PD;
    float* h1   = ws;  ws += B_ * PD;
    float* yimp = ws;  ws += B_;
    float* hs   = ws;  ws += B_ * PD;
    float* ysym = ws;  ws += B_;
    float* out  = (float*)d_out;

    k_init      <<<N_ / 256, 256, 0, stream>>>(deg, sums, cnts);
    k_deg       <<<E_ / 256, 256, 0, stream>>>(ei, deg);
    k_selfloop  <<<(int)(((size_t)N_ * PD) / 256), 256, 0, stream>>>(x, deg, agg);
    k_edges     <<<E_ / 8, 256, 0, stream>>>(ei, x, deg, agg);
    k_cnts      <<<N_ / 256, 256, 0, stream>>>(bv, cnts);
    k_gcn_gemm_pool<<<(N_ / 16) * (PD / 16) / 8, 256, 0, stream>>>(agg, Wg, bg, bv, sums);
    k_proj      <<<B_ * 1024 / 256, 256, 0, stream>>>(imgF, txtF, Wi, bi, Wt, bt, comb);
    k_h1        <<<B_ * PD / 256, 256, 0, stream>>>(comb, W1, b1, h1);
    k_vec1      <<<1, 128, 0, stream>>>(h1, W2, b2, yimp);
    k_hs        <<<B_ * PD / 256, 256, 0, stream>>>(sums, cnts, Ws1, bs1, hs);
    k_vec1      <<<1, 128, 0, stream>>>(hs, Ws2, bs2, ysym);
    k_final     <<<1, 128, 0, stream>>>(yimp, ysym, ans, out);
}